// T2T_module_17763984736763
// MI455X (gfx1250) — compile-verified
//
#include <hip/hip_runtime.h>
#include <math.h>

// ---------------------------------------------------------------------------
// CDNA5 (gfx1250) wave32 WMMA f32 16x16x4 helpers.
// Layouts (ISA 7.12.2):
//   A (16x4):  lane L holds row m=L%16, K pair kb=2*(L/16)  -> {A[m][kb],A[m][kb+1]}
//   B (4x16):  lane L holds col n=L%16, K pair kb=2*(L/16)  -> {B[kb][n],B[kb+1][n]}
//   C/D (16x16): VGPR r, half h=L/16 -> element [r+8h][L%16]
// ---------------------------------------------------------------------------
typedef __attribute__((ext_vector_type(2))) float v2f;
typedef __attribute__((ext_vector_type(8))) float v8f;

__device__ __forceinline__ v8f wmma4(v2f a, v2f b, v8f c) {
  return __builtin_amdgcn_wmma_f32_16x16x4_f32(false, a, false, b, (short)0, c,
                                               false, false);
}

__device__ __forceinline__ float wave_sum32(float v) {
#pragma unroll
  for (int off = 16; off > 0; off >>= 1) v += __shfl_xor(v, off, 32);
  return v;
}

__device__ __forceinline__ int shift_dh(int g) { return (g == 0) ? 0 : (g <= 2 ? -1 : 1); }
__device__ __forceinline__ int shift_dw(int g) { return (g == 0) ? 0 : ((g & 1) ? -1 : 1); }

// ---------------------------------------------------------------------------
// Weight repack: WT[n][kpad] = (k < K) ? W[k][n] : 0.
// Makes B fragments contiguous (b64 loads) and removes all K-guards.
// ---------------------------------------------------------------------------
__global__ void transpose_pad(const float* __restrict__ W, float* __restrict__ WT,
                              int K, int N, int Kpad, int total) {
  const int idx = blockIdx.x * blockDim.x + threadIdx.x;
  if (idx >= total) return;
  const int n = idx / Kpad, k = idx % Kpad;
  WT[idx] = (k < K) ? W[(size_t)k * N + n] : 0.f;
}

// ---------------------------------------------------------------------------
// Generic WMMA GEMM:  C[M,N] = act(A[M,Kpad] @ WT^T + bias) + res
// Wave -> 16x32 C tile (2 N-tiles reuse each A fragment).
// 4 independent accumulator chains hide WMMA RAW hazard. Kpad % 8 == 0.
// ---------------------------------------------------------------------------
__global__ __launch_bounds__(128) void gemm_wmma(
    const float* __restrict__ A, int lda, const float* __restrict__ WT, int ldw,
    const float* __restrict__ bias, const float* __restrict__ res, int ldres,
    int resoff, float* __restrict__ C, int ldc, int K, int gelu) {
  const int lane = threadIdx.x & 31;
  const int w = threadIdx.x >> 5;
  const int row0 = (blockIdx.y * 4 + w) * 16;
  const int col0 = blockIdx.x * 32;
  const int m = lane & 15;
  const int kb = (lane >> 4) * 2;
  const int half = lane >> 4;
  const float* Ap = A + (size_t)(row0 + m) * lda + kb;
  const float* B0 = WT + (size_t)(col0 + m) * ldw + kb;
  const float* B1 = WT + (size_t)(col0 + 16 + m) * ldw + kb;

  v8f a00 = {}, a01 = {}, a10 = {}, a11 = {};
  for (int k0 = 0; k0 < K; k0 += 8) {
    const v2f af0 = *(const v2f*)(Ap + k0);
    const v2f af1 = *(const v2f*)(Ap + k0 + 4);
    const v2f b00 = *(const v2f*)(B0 + k0);
    const v2f b01 = *(const v2f*)(B1 + k0);
    const v2f b10 = *(const v2f*)(B0 + k0 + 4);
    const v2f b11 = *(const v2f*)(B1 + k0 + 4);
    a00 = wmma4(af0, b00, a00);
    a01 = wmma4(af0, b01, a01);
    a10 = wmma4(af1, b10, a10);
    a11 = wmma4(af1, b11, a11);
  }
  const v8f sum0 = a00 + a10;
  const v8f sum1 = a01 + a11;
#pragma unroll
  for (int t = 0; t < 2; ++t) {
    const v8f acc = t ? sum1 : sum0;
#pragma unroll
    for (int r = 0; r < 8; ++r) {
      const int rr = row0 + r + 8 * half;
      const int cc = col0 + t * 16 + m;
      float v = acc[r];
      if (bias) v += bias[cc];
      if (gelu) v = 0.5f * v * (1.f + erff(v * 0.70710678118654752f));
      if (res) v += res[(size_t)rr * ldres + resoff + cc];
      C[(size_t)rr * ldc + cc] = v;
    }
  }
}

// ---------------------------------------------------------------------------
// Stage-1 gather: patch_shift + unfold3x3(s2,p1) + LayerNorm(135) -> H (ld 136,
// col 135 zeroed). One wave per token (B*1024 tokens).
// ---------------------------------------------------------------------------
__global__ __launch_bounds__(128) void g1_kernel(const float* __restrict__ x,
                                                 const float* __restrict__ gam,
                                                 const float* __restrict__ bet,
                                                 float* __restrict__ H) {
  const int lane = threadIdx.x & 31;
  const int w = threadIdx.x >> 5;
  const int token = blockIdx.x * 4 + w;  // b*1024 + n
  const int b = token >> 10, n = token & 1023;
  const int oh = n >> 5, ow = n & 31;
  float vals[5], s = 0.f, s2 = 0.f;
#pragma unroll
  for (int t = 0; t < 5; ++t) {
    const int c = lane + 32 * t;
    float v = 0.f;
    if (c < 135) {
      const int ci = c / 9, kidx = c % 9;
      const int g = ci / 3, ch = ci % 3;
      const int i = kidx / 3, j = kidx % 3;
      const int h = 2 * oh + i - 1 + shift_dh(g);
      const int wc = 2 * ow + j - 1 + shift_dw(g);
      if (h >= 0 && h < 64 && wc >= 0 && wc < 64)
        v = x[(((size_t)b * 3 + ch) * 64 + h) * 64 + wc];
    }
    vals[t] = v;
    s += v;
    s2 += v * v;
  }
  s = wave_sum32(s);
  s2 = wave_sum32(s2);
  const float mean = s * (1.f / 135.f);
  const float var = s2 * (1.f / 135.f) - mean * mean;
  const float rstd = rsqrtf(var + 1e-5f);
#pragma unroll
  for (int t = 0; t < 5; ++t) {
    const int c = lane + 32 * t;
    if (c < 135)
      H[(size_t)token * 136 + c] = (vals[t] - mean) * rstd * gam[c] + bet[c];
  }
  if (lane == 0) H[(size_t)token * 136 + 135] = 0.f;
}

// ---------------------------------------------------------------------------
// Stage-2 gather: reshape (B,1024,64)->(B,64,32,32), unfold3x3(s2,p1),
// LayerNorm(576). One wave per token (B*256), 18 elems/lane (18*32 == 576).
// ---------------------------------------------------------------------------
__global__ __launch_bounds__(128) void g2_kernel(const float* __restrict__ X1,
                                                 const float* __restrict__ gam,
                                                 const float* __restrict__ bet,
                                                 float* __restrict__ H) {
  const int lane = threadIdx.x & 31;
  const int w = threadIdx.x >> 5;
  const int token = blockIdx.x * 4 + w;  // b*256 + n
  const int b = token >> 8, n = token & 255;
  const int oh = n >> 4, ow = n & 15;
  float vals[18], s = 0.f, s2 = 0.f;
#pragma unroll
  for (int t = 0; t < 18; ++t) {
    const int c = lane + 32 * t;
    const int ch = c / 9, kidx = c % 9;
    const int i = kidx / 3, j = kidx % 3;
    const int h = 2 * oh + i - 1;
    const int wc = 2 * ow + j - 1;
    float v = 0.f;
    if (h >= 0 && h < 32 && wc >= 0 && wc < 32)
      v = X1[((size_t)b * 1024 + h * 32 + wc) * 64 + ch];
    vals[t] = v;
    s += v;
    s2 += v * v;
  }
  s = wave_sum32(s);
  s2 = wave_sum32(s2);
  const float mean = s * (1.f / 576.f);
  const float var = s2 * (1.f / 576.f) - mean * mean;
  const float rstd = rsqrtf(var + 1e-5f);
#pragma unroll
  for (int t = 0; t < 18; ++t) {
    const int c = lane + 32 * t;
    H[(size_t)token * 576 + c] = (vals[t] - mean) * rstd * gam[c] + bet[c];
  }
}

// ---------------------------------------------------------------------------
// LayerNorm over 64 dims, one wave per token.
// ---------------------------------------------------------------------------
__global__ __launch_bounds__(128) void ln_kernel(const float* __restrict__ X,
                                                 const float* __restrict__ gam,
                                                 const float* __restrict__ bet,
                                                 float* __restrict__ Y) {
  const int lane = threadIdx.x & 31;
  const int w = threadIdx.x >> 5;
  const int token = blockIdx.x * 4 + w;
  const float v0 = X[(size_t)token * 64 + lane];
  const float v1 = X[(size_t)token * 64 + lane + 32];
  float s = wave_sum32(v0 + v1);
  float s2 = wave_sum32(v0 * v0 + v1 * v1);
  const float mean = s * (1.f / 64.f);
  const float var = s2 * (1.f / 64.f) - mean * mean;
  const float rstd = rsqrtf(var + 1e-5f);
  Y[(size_t)token * 64 + lane] = (v0 - mean) * rstd * gam[lane] + bet[lane];
  Y[(size_t)token * 64 + lane + 32] =
      (v1 - mean) * rstd * gam[lane + 32] + bet[lane + 32];
}

// ---------------------------------------------------------------------------
// Stage-3 gather: reshape (B,256,64)->(B,64,16,16), patch_shift (320 ch),
// unfold3x3(s2,p1) -> X3 (B*64, 2880). Elementwise.
// ---------------------------------------------------------------------------
__global__ void g3_kernel(const float* __restrict__ X2, float* __restrict__ X3,
                          int total) {
  const int idx = blockIdx.x * blockDim.x + threadIdx.x;
  if (idx >= total) return;
  const int c = idx % 2880;
  const int t = idx / 2880;
  const int n = t % 64, b = t / 64;
  const int oh = n >> 3, ow = n & 7;
  const int cg = c / 9, kidx = c % 9;
  const int g = cg >> 6, ch = cg & 63;
  const int i = kidx / 3, j = kidx % 3;
  const int h = 2 * oh + i - 1 + shift_dh(g);
  const int wc = 2 * ow + j - 1 + shift_dw(g);
  float v = 0.f;
  if (h >= 0 && h < 16 && wc >= 0 && wc < 16)
    v = X2[((size_t)b * 256 + h * 16 + wc) * 64 + ch];
  X3[idx] = v;
}

// ---------------------------------------------------------------------------
// Flash attention (diag-masked), d=64, runtime N (1024 / 256).
// Block = 4 waves = 64 queries. K/V tiles of 32 keys in LDS; S and P@V via
// WMMA; P staged through a per-wave LDS tile for C->A layout conversion;
// online softmax via 16-lane shuffle reductions.
// ---------------------------------------------------------------------------
__global__ __launch_bounds__(128) void attn_kernel(
    const float* __restrict__ QKV, float* __restrict__ O,
    const float* __restrict__ scale_p, int N) {
  __shared__ float Ks[32][68];
  __shared__ float Vs[32][68];
  __shared__ float Ps[4][16][36];
  const int lane = threadIdx.x & 31;
  const int w = threadIdx.x >> 5;
  const int b = blockIdx.x;
  const int q0 = blockIdx.y * 64 + w * 16;
  const int m = lane & 15;
  const int kb = (lane >> 4) * 2;
  const int half = lane >> 4;
  const float scale = scale_p[0];

  // Preload Q fragments (16 k-steps of 4), pre-scaled; b64 loads.
  v2f qa[16];
  {
    const float* qrow = QKV + ((size_t)b * N + q0 + m) * 192 + kb;
#pragma unroll
    for (int ks = 0; ks < 16; ++ks) {
      v2f q = *(const v2f*)(qrow + ks * 4);
      qa[ks].x = q.x * scale;
      qa[ks].y = q.y * scale;
    }
  }
  v8f Oa[4] = {};
  float mrun[8], lrun[8];
#pragma unroll
  for (int r = 0; r < 8; ++r) {
    mrun[r] = -__builtin_inff();
    lrun[r] = 0.f;
  }
  const int ntiles = N >> 5;
  for (int kt = 0; kt < ntiles; ++kt) {
    for (int idx = threadIdx.x; idx < 32 * 32; idx += 128) {
      const int r = idx >> 5, cc = (idx & 31) * 2;
      const float* src = QKV + ((size_t)b * N + kt * 32 + r) * 192;
      *(v2f*)(&Ks[r][cc]) = *(const v2f*)(src + 64 + cc);
      *(v2f*)(&Vs[r][cc]) = *(const v2f*)(src + 128 + cc);
    }
    __syncthreads();
    v8f S[2];
#pragma unroll
    for (int sub = 0; sub < 2; ++sub) {
      v8f s = {};
#pragma unroll
      for (int ks = 0; ks < 16; ++ks) {
        const v2f bf = *(const v2f*)(&Ks[sub * 16 + m][ks * 4 + kb]);
        s = wmma4(qa[ks], bf, s);
      }
      const int kcol = kt * 32 + sub * 16 + m;
#pragma unroll
      for (int r = 0; r < 8; ++r)
        if (kcol == q0 + r + 8 * half) s[r] = -__builtin_inff();
      S[sub] = s;
    }
    // Online softmax per row (rows live in 16-lane halves).
    float alpha[8];
#pragma unroll
    for (int r = 0; r < 8; ++r) {
      float tmax = fmaxf(S[0][r], S[1][r]);
#pragma unroll
      for (int off = 1; off < 16; off <<= 1)
        tmax = fmaxf(tmax, __shfl_xor(tmax, off, 32));
      const float mnew = fmaxf(mrun[r], tmax);
      alpha[r] = __expf(mrun[r] - mnew);
      const float p0 = __expf(S[0][r] - mnew);
      const float p1 = __expf(S[1][r] - mnew);
      S[0][r] = p0;
      S[1][r] = p1;
      float ps = p0 + p1;
#pragma unroll
      for (int off = 1; off < 16; off <<= 1) ps += __shfl_xor(ps, off, 32);
      lrun[r] = lrun[r] * alpha[r] + ps;
      mrun[r] = mnew;
    }
#pragma unroll
    for (int nt = 0; nt < 4; ++nt)
#pragma unroll
      for (int r = 0; r < 8; ++r) Oa[nt][r] *= alpha[r];
    // C-layout -> A-layout via per-wave LDS tile.
#pragma unroll
    for (int sub = 0; sub < 2; ++sub)
#pragma unroll
      for (int r = 0; r < 8; ++r)
        Ps[w][r + 8 * half][sub * 16 + m] = S[sub][r];
    // O += P(16x32) @ V(32x64)
#pragma unroll
    for (int ks2 = 0; ks2 < 8; ++ks2) {
      const v2f af = *(const v2f*)(&Ps[w][m][ks2 * 4 + kb]);
#pragma unroll
      for (int nt = 0; nt < 4; ++nt) {
        v2f bf;
        bf.x = Vs[ks2 * 4 + kb][nt * 16 + m];
        bf.y = Vs[ks2 * 4 + kb + 1][nt * 16 + m];
        Oa[nt] = wmma4(af, bf, Oa[nt]);
      }
    }
    __syncthreads();
  }
#pragma unroll
  for (int nt = 0; nt < 4; ++nt)
#pragma unroll
    for (int r = 0; r < 8; ++r) {
      const int qrow_i = q0 + r + 8 * half;
      O[((size_t)b * N + qrow_i) * 64 + nt * 16 + m] = Oa[nt][r] / lrun[r];
    }
}

// ---------------------------------------------------------------------------
static void launch_gemm(const float* A, int lda, const float* WT, int ldw,
                        const float* bias, const float* res, int ldres,
                        int resoff, float* C, int ldc, int M, int N, int K,
                        int gelu, hipStream_t s) {
  dim3 grid(N / 32, M / 64);
  gemm_wmma<<<grid, 128, 0, s>>>(A, lda, WT, ldw, bias, res, ldres, resoff, C,
                                 ldc, K, gelu);
}

static void launch_tpad(const float* W, float* WT, int K, int N, int Kpad,
                        hipStream_t s) {
  const int total = N * Kpad;
  transpose_pad<<<(total + 255) / 256, 256, 0, s>>>(W, WT, K, N, Kpad, total);
}

extern "C" void kernel_launch(void* const* d_in, const int* in_sizes, int n_in,
                              void* d_out, int out_size, void* d_ws,
                              size_t ws_size, hipStream_t stream) {
  (void)in_sizes; (void)n_in; (void)out_size; (void)ws_size;
  const float* x = (const float*)d_in[0];
  const float* p1[12];
  const float* p2[12];
  for (int i = 0; i < 12; ++i) p1[i] = (const float*)d_in[1 + i];
  for (int i = 0; i < 12; ++i) p2[i] = (const float*)d_in[13 + i];
  // tf param order: ln1_g, ln1_b, qkv_w, scale, proj_w, proj_b, ln2_g, ln2_b,
  //                 fc1_w, fc1_b, fc2_w, fc2_b
  const float* projw = (const float*)d_in[25];
  const float* projb = (const float*)d_in[26];

  float* ws = (float*)d_ws;
  float* H = ws;                    // 11,796,480 floats (H0/H2/X3)
  float* QKV = H + 11796480;        // 12,582,912
  float* Ob = QKV + 12582912;       // 4,194,304
  float* X1 = Ob + 4194304;         // 4,194,304 (residual x1)
  float* HL = X1 + 4194304;         // 4,194,304 (ln2 out / stage2 out)
  float* G = HL + 4194304;          // 4,194,304 (gelu hidden)
  float* OUT1 = G + 4194304;        // 4,194,304 (stage1 output)
  float* qkv1T = OUT1 + 4194304;    // 192*136
  float* qkv2T = qkv1T + 192 * 136; // 192*576
  float* pr1T = qkv2T + 192 * 576;  // 64*64 each
  float* f11T = pr1T + 4096;
  float* f21T = f11T + 4096;
  float* pr2T = f21T + 4096;
  float* f12T = pr2T + 4096;
  float* f22T = f12T + 4096;
  float* projT = f22T + 4096;       // 768*2880

  // Repack all weights (transposed + K zero-padded).
  launch_tpad(p1[2], qkv1T, 135, 192, 136, stream);
  launch_tpad(p1[4], pr1T, 64, 64, 64, stream);
  launch_tpad(p1[8], f11T, 64, 64, 64, stream);
  launch_tpad(p1[10], f21T, 64, 64, 64, stream);
  launch_tpad(p2[2], qkv2T, 576, 192, 576, stream);
  launch_tpad(p2[4], pr2T, 64, 64, 64, stream);
  launch_tpad(p2[8], f12T, 64, 64, 64, stream);
  launch_tpad(p2[10], f22T, 64, 64, 64, stream);
  launch_tpad(projw, projT, 2880, 768, 2880, stream);

  // ---------------- stage 1 (N=1024, dim=135 -> Kpad 136) ----------------
  const int M1 = 64 * 1024;
  g1_kernel<<<M1 / 4, 128, 0, stream>>>(x, p1[0], p1[1], H);
  launch_gemm(H, 136, qkv1T, 136, nullptr, nullptr, 0, 0, QKV, 192, M1, 192,
              136, 0, stream);
  {
    dim3 ag(64, 1024 / 64);
    attn_kernel<<<ag, 128, 0, stream>>>(QKV, Ob, p1[3], 1024);
  }
  launch_gemm(Ob, 64, pr1T, 64, p1[5], QKV, 192, 128, X1, 64, M1, 64, 64, 0,
              stream);  // x1 = v + (a@proj + b)
  ln_kernel<<<M1 / 4, 128, 0, stream>>>(X1, p1[6], p1[7], HL);
  launch_gemm(HL, 64, f11T, 64, p1[9], nullptr, 0, 0, G, 64, M1, 64, 64, 1,
              stream);  // gelu(h@fc1 + b)
  launch_gemm(G, 64, f21T, 64, p1[11], X1, 64, 0, OUT1, 64, M1, 64, 64, 0,
              stream);  // x1 + (g@fc2 + b)

  // ---------------- stage 2 (N=256, dim=576) ----------------
  const int M2 = 64 * 256;
  g2_kernel<<<M2 / 4, 128, 0, stream>>>(OUT1, p2[0], p2[1], H);
  launch_gemm(H, 576, qkv2T, 576, nullptr, nullptr, 0, 0, QKV, 192, M2, 192,
              576, 0, stream);
  {
    dim3 ag(64, 256 / 64);
    attn_kernel<<<ag, 128, 0, stream>>>(QKV, Ob, p2[3], 256);
  }
  launch_gemm(Ob, 64, pr2T, 64, p2[5], QKV, 192, 128, X1, 64, M2, 64, 64, 0,
              stream);
  ln_kernel<<<M2 / 4, 128, 0, stream>>>(X1, p2[6], p2[7], HL);
  launch_gemm(HL, 64, f12T, 64, p2[9], nullptr, 0, 0, G, 64, M2, 64, 64, 1,
              stream);
  launch_gemm(G, 64, f22T, 64, p2[11], X1, 64, 0, HL, 64, M2, 64, 64, 0,
              stream);  // stage-2 output in HL

  // ---------------- stage 3: patch_shift + unfold + final proj ------------
  const int tot3 = 64 * 64 * 2880;
  g3_kernel<<<(tot3 + 255) / 256, 256, 0, stream>>>(HL, H, tot3);
  launch_gemm(H, 2880, projT, 2880, projb, nullptr, 0, 0, (float*)d_out, 768,
              64 * 64, 768, 2880, 0, stream);
}